// DilatedAttention_56573309223859
// MI455X (gfx1250) — compile-verified
//
#include <hip/hip_runtime.h>

typedef float v2f __attribute__((ext_vector_type(2)));
typedef float v4f __attribute__((ext_vector_type(4)));
typedef float v8f __attribute__((ext_vector_type(8)));
typedef int   v4i_g __attribute__((vector_size(16)));   // matches builtin param type

#define HEADS    16
#define SEQ      4096
#define DIM      64
#define RATE     2
#define NSEQ     2048          // SEQ / RATE (per-offset dilated length)
#define NTILES   128           // NSEQ / 16
#define HALF_WIN 64
#define TW       9             // key tiles covering the band per query tile
#define PSTR     146           // ldsP row stride (dwords), even + bank-spread
#define VSTR     68            // ldsV row stride (dwords), 16B-aligned rows

#if __has_builtin(__builtin_amdgcn_global_load_async_to_lds_b128)
#define HAVE_ASYNC_LDS 1
#endif

__launch_bounds__(32)
__global__ void dilated_attn_f32_wmma(const float* __restrict__ q,
                                      const float* __restrict__ k,
                                      const float* __restrict__ v,
                                      float* __restrict__ out) {
  __shared__ float ldsP[16 * PSTR];     // P rows:  [query i][band j]
  __shared__ float ldsV[TW * 16 * VSTR];// V tiles: [band j][d] (row-major)

  const int lane = threadIdx.x;
  const int row  = lane & 15;
  const int hi   = lane >> 4;           // 0: lanes 0-15, 1: lanes 16-31

  const int t   = blockIdx.x;           // query tile index in dilated seq
  const int hr  = blockIdx.y;           // h * RATE + offset
  const int h   = hr >> 1;
  const int off = hr & 1;

  const int i0 = t * 16;
  const int iq = i0 + row;              // this lane's query (column of S^T)
  const long headBase = (long)h * SEQ * DIM;

  // ---- Q in WMMA operand layout: lane holds M/N=row, Kdim = 2*hi+{0,1} ----
  const float* qrow = q + headBase + (long)(off + RATE * iq) * DIM + 2 * hi;
  v2f qa[16];
#pragma unroll
  for (int kk = 0; kk < 16; ++kk)
    qa[kk] = *(const v2f*)(qrow + 4 * kk);

  // ---- stage V tiles into LDS row-major via async copy (ASYNCcnt) ----
#pragma unroll 1
  for (int dj = 0; dj < TW; ++dj) {
    const int jt = t + dj - 4;
    if (jt < 0 || jt >= NTILES) continue;
    const int j0 = jt * 16;
#pragma unroll
    for (int it = 0; it < 8; ++it) {
      const int jr = 2 * it + hi;                       // row within key tile
      const int d4 = row * 4;                           // 16B chunk per lane
      const float* g = v + headBase + (long)(off + RATE * (j0 + jr)) * DIM + d4;
      const int jb = dj * 16 + jr;
#if defined(HAVE_ASYNC_LDS)
      // LDS byte offset = low 32 bits of the generic LDS pointer (aperture map)
      __builtin_amdgcn_global_load_async_to_lds_b128(
          (v4i_g*)g,
          (__attribute__((address_space(3))) v4i_g*)(unsigned int)(unsigned long long)
              &ldsV[jb * VSTR + d4],
          0, 0);
#else
      *(v4f*)&ldsV[jb * VSTR + d4] = *(const v4f*)g;
#endif
    }
  }

  // ---- Pass 1: S^T tiles = K @ Q^T via v_wmma_f32_16x16x4_f32 ----
  float sc[TW][8];
#pragma unroll
  for (int dj = 0; dj < TW; ++dj) {
    const int jt = t + dj - 4;
    if (jt < 0 || jt >= NTILES) {
#pragma unroll
      for (int p = 0; p < 8; ++p) sc[dj][p] = -__builtin_huge_valf();
      continue;
    }
    const int j0 = jt * 16;
    const float* krow = k + headBase + (long)(off + RATE * (j0 + row)) * DIM + 2 * hi;
    v8f c = {};
#pragma unroll
    for (int kk = 0; kk < 16; ++kk) {
      v2f kb = *(const v2f*)(krow + 4 * kk);
      // A = K-tile (16 keys x 4 d), B = Q^T (4 d x 16 queries)
      c = __builtin_amdgcn_wmma_f32_16x16x4_f32(false, kb, false, qa[kk],
                                                (short)0, c, false, false);
    }
#pragma unroll
    for (int p = 0; p < 8; ++p) {
      const int jg = j0 + p + 8 * hi;   // key index (row M of S^T)
      const bool inband = (jg >= iq - HALF_WIN) && (jg <= iq + HALF_WIN);
      sc[dj][p] = inband ? c[p] : -__builtin_huge_valf();
    }
  }

  // ---- softmax per query: in-lane reduce + one xor-16 shuffle ----
  float m = -__builtin_huge_valf();
#pragma unroll
  for (int dj = 0; dj < TW; ++dj)
#pragma unroll
    for (int p = 0; p < 8; ++p) m = fmaxf(m, sc[dj][p]);
  m = fmaxf(m, __shfl_xor(m, 16, 32));

  const float cexp = 0.125f * 1.44269504088896340736f;  // 1/sqrt(64) * log2(e)
  float l = 0.0f;
#pragma unroll
  for (int dj = 0; dj < TW; ++dj)
#pragma unroll
    for (int p = 0; p < 8; ++p) {
      const float e = exp2f((sc[dj][p] - m) * cexp);
      sc[dj][p] = e;
      l += e;
    }
  l += __shfl_xor(l, 16, 32);
  const float inv = 1.0f / l;           // applied to O at the end

  // ---- write P rows to LDS: ldsP[i][band_j] (lane owns query i = row) ----
#pragma unroll
  for (int dj = 0; dj < TW; ++dj)
#pragma unroll
    for (int p2 = 0; p2 < 4; ++p2) {
      v2f e2 = { sc[dj][2 * p2], sc[dj][2 * p2 + 1] };
      *(v2f*)&ldsP[row * PSTR + dj * 16 + 8 * hi + 2 * p2] = e2;
    }

#if defined(HAVE_ASYNC_LDS)
#if __has_builtin(__builtin_amdgcn_s_wait_asynccnt)
  __builtin_amdgcn_s_wait_asynccnt(0);
#else
  asm volatile("s_wait_asynccnt 0x0" ::: "memory");
#endif
#endif
  __syncthreads();

  // ---- Pass 2: O^T = V^T @ P^T (queries stay per-lane) ----
  const int dlo = (t < 4) ? (4 - t) : 0;
  const int dhi = (t + 4 >= NTILES) ? (NTILES - 1 - t + 4) : 8;
  float* orow = out + headBase + (long)(off + RATE * iq) * DIM + 8 * hi;
#pragma unroll
  for (int db = 0; db < 4; ++db) {
    const int d0 = db * 16;
    v8f c = {};
    for (int dj = dlo; dj <= dhi; ++dj) {
#pragma unroll
      for (int js = 0; js < 4; ++js) {
        const int jb = dj * 16 + 4 * js + 2 * hi;
        v2f a;                                           // A = V^T tile
        a.x = ldsV[(jb + 0) * VSTR + d0 + row];
        a.y = ldsV[(jb + 1) * VSTR + d0 + row];
        v2f b = *(const v2f*)&ldsP[row * PSTR + jb];     // B = P^T tile
        c = __builtin_amdgcn_wmma_f32_16x16x4_f32(false, a, false, b,
                                                  (short)0, c, false, false);
      }
    }
    v4f o0, o1;
#pragma unroll
    for (int p = 0; p < 4; ++p) { o0[p] = c[p] * inv; o1[p] = c[p + 4] * inv; }
    *(v4f*)(orow + d0)     = o0;   // d = d0 + 8*hi + 0..3
    *(v4f*)(orow + d0 + 4) = o1;   // d = d0 + 8*hi + 4..7
  }
}

extern "C" void kernel_launch(void* const* d_in, const int* in_sizes, int n_in,
                              void* d_out, int out_size, void* d_ws, size_t ws_size,
                              hipStream_t stream) {
  const float* q = (const float*)d_in[0];
  const float* k = (const float*)d_in[1];
  const float* v = (const float*)d_in[2];
  float* o = (float*)d_out;
  (void)in_sizes; (void)n_in; (void)out_size; (void)d_ws; (void)ws_size;

  dim3 grid(NTILES, HEADS * RATE);  // 128 query tiles x (16 heads * 2 offsets)
  dim3 block(32);                   // one wave32 per query tile
  dilated_attn_f32_wmma<<<grid, block, 0, stream>>>(q, k, v, o);
}